// Net_39419209843103
// MI455X (gfx1250) — compile-verified
//
#include <hip/hip_runtime.h>
#include <hip/hip_bf16.h>

// CDNA5 / gfx1250, wave32.
// loss[e] = -( log_sigmoid(dot(u,v)) + sum_k log_sigmoid(-dot(nu_k, nv_k)) )
// One wave handles 16 edges. Each of the 6 dot-product "chains" (1 pos + 5 neg)
// is computed as diag( U(16x128) * V^T(128x16) ) via 32 chained
// V_WMMA_F32_16X16X4_F32 ops. Memory-bound: ~300MB of 512B row gathers.

typedef float v2f __attribute__((ext_vector_type(2)));
typedef float v8f __attribute__((ext_vector_type(8)));

#define DIM  128
#define NEGK 5

__device__ __forceinline__ float log_sigmoid(float x) {
    // stable: min(x,0) - log1p(exp(-|x|))
    return fminf(x, 0.0f) - log1pf(__expf(-fabsf(x)));
}

// C/D 16x16 f32 layout: VGPR r: lanes 0-15 -> (M=r, N=lane), lanes 16-31 -> (M=r+8, N=lane-16).
// diag[m]: m<8 at (vgpr m, lane m); m>=8 at (vgpr m-8, lane m+16).
__device__ __forceinline__ float diag_extract(v8f acc, int lane) {
    int r = lane & 7;
    float x = acc[0];
    x = (r == 1) ? acc[1] : x;
    x = (r == 2) ? acc[2] : x;
    x = (r == 3) ? acc[3] : x;
    x = (r == 4) ? acc[4] : x;
    x = (r == 5) ? acc[5] : x;
    x = (r == 6) ? acc[6] : x;
    x = (r == 7) ? acc[7] : x;
    int m   = lane & 15;
    int src = (m < 8) ? m : (m + 16);   // source lane holding diag[m]
    return __shfl(x, src, 32);
}

__global__ __launch_bounds__(256)
void n2v_loss_kernel(const float* __restrict__ emb,
                     const int*   __restrict__ pos_edges,   // [E][2]
                     const int*   __restrict__ neg_edges,   // [E][K][2]
                     float*       __restrict__ out,         // [E]
                     int E)
{
    const int lane = threadIdx.x & 31;
    const int wid  = blockIdx.x * (blockDim.x >> 5) + (threadIdx.x >> 5);
    const int e0   = wid * 16;
    if (e0 >= E) return;                 // wave-uniform exit, EXEC stays full

    const int  m     = lane & 15;        // which of the 16 edges this lane serves
    const int  h     = lane >> 4;        // K-half: lanes 16-31 supply K+2,K+3
    int        e     = e0 + m;
    const bool valid = (e < E) && (lane < 16);
    if (e >= E) e = E - 1;               // clamp; keep all lanes live for WMMA

    // Row base pointers (pre-offset by this lane's K-half) for 6 chains.
    const float* ru[6];
    const float* rv[6];
    {
        int iu = pos_edges[2 * e + 0];
        int iv = pos_edges[2 * e + 1];
        ru[0] = emb + (size_t)iu * DIM + 2 * h;
        rv[0] = emb + (size_t)iv * DIM + 2 * h;
    }
#pragma unroll
    for (int k = 0; k < NEGK; ++k) {
        int iu = neg_edges[((size_t)e * NEGK + k) * 2 + 0];
        int iv = neg_edges[((size_t)e * NEGK + k) * 2 + 1];
        ru[1 + k] = emb + (size_t)iu * DIM + 2 * h;
        rv[1 + k] = emb + (size_t)iv * DIM + 2 * h;
    }

    // Get the first cacheline of every row moving early (global_prefetch_b8).
#pragma unroll
    for (int t = 0; t < 6; ++t) {
        __builtin_prefetch(ru[t], 0, 0);
        __builtin_prefetch(rv[t], 0, 0);
    }

    float dots[6];
#pragma unroll
    for (int t = 0; t < 6; ++t) {
        v8f acc = {};
#pragma unroll
        for (int k0 = 0; k0 < DIM; k0 += 4) {
            // A 16x4: lane -> A[m][k0 + 2h + {0,1}] ; B 4x16: lane -> B[k0 + 2h + {0,1}][m]
            v2f a = *(const v2f*)(ru[t] + k0);
            v2f b = *(const v2f*)(rv[t] + k0);
            acc = __builtin_amdgcn_wmma_f32_16x16x4_f32(
                /*neg_a=*/false, a, /*neg_b=*/false, b,
                /*c_mod=*/(short)0, acc, /*reuse_a=*/false, /*reuse_b=*/false);
        }
        dots[t] = diag_extract(acc, lane);
    }

    float s = log_sigmoid(dots[0]);
#pragma unroll
    for (int k = 0; k < NEGK; ++k) s += log_sigmoid(-dots[1 + k]);

    if (valid) out[e0 + m] = -s;
}

extern "C" void kernel_launch(void* const* d_in, const int* in_sizes, int n_in,
                              void* d_out, int out_size, void* d_ws, size_t ws_size,
                              hipStream_t stream) {
    const float* emb = (const float*)d_in[0];
    const int*   pos = (const int*)d_in[1];
    const int*   neg = (const int*)d_in[2];
    float*       out = (float*)d_out;

    const int E = in_sizes[1] / 2;                  // pos_edges is [E][2]
    const int waves          = (E + 15) / 16;       // 16 edges per wave
    const int wavesPerBlock  = 8;                   // 256 threads
    const int blocks         = (waves + wavesPerBlock - 1) / wavesPerBlock;

    n2v_loss_kernel<<<blocks, 256, 0, stream>>>(emb, pos, neg, out, E);
}